// APJFFF_text_49228915146813
// MI455X (gfx1250) — compile-verified
//
#include <hip/hip_runtime.h>
#include <cmath>

typedef float v2f __attribute__((ext_vector_type(2)));
typedef float v8f __attribute__((ext_vector_type(8)));

#define B_   16
#define L_   128
#define DIN_ 768
#define HD_  128
#define G_   512   // 4*HD
#define D_   256   // 2*HD
#define HS_LD 132  // padded LDS stride for h

__device__ __forceinline__ float sigmf(float x) { return 1.0f / (1.0f + expf(-x)); }

// ---------------------------------------------------------------------------
// Generic fp32 WMMA GEMM: C[M,N] = act(A[M,K] @ W[N,K]^T + b1[N] + b2[N])
// grid = (N/16, M/16), block = 32 (one wave per 16x16 output tile).
// ACT: 0=none, 1=tanh, 2=relu
// ---------------------------------------------------------------------------
template <int ACT>
__global__ __launch_bounds__(32) void gemm_wmma(
    const float* __restrict__ A, const float* __restrict__ W,
    const float* __restrict__ b1, const float* __restrict__ b2,
    float* __restrict__ C, int N, int K) {
  const int lane = threadIdx.x;
  const int l16 = lane & 15;
  const int half = lane >> 4;
  const int n0 = blockIdx.x * 16;
  const int m0 = blockIdx.y * 16;

  const float* arow = A + (size_t)(m0 + l16) * K + half * 2;
  const float* brow = W + (size_t)(n0 + l16) * K + half * 2;

  v8f acc = {0.f, 0.f, 0.f, 0.f, 0.f, 0.f, 0.f, 0.f};
  for (int k = 0; k < K; k += 4) {
    v2f av = *(const v2f*)(arow + k);
    v2f bv = *(const v2f*)(brow + k);
    acc = __builtin_amdgcn_wmma_f32_16x16x4_f32(false, av, false, bv,
                                                (short)0, acc, false, false);
  }

  float bias = 0.0f;
  if (b1) bias += b1[n0 + l16];
  if (b2) bias += b2[n0 + l16];

#pragma unroll
  for (int j = 0; j < 8; j++) {
    float v = acc[j] + bias;
    if (ACT == 1) v = tanhf(v);
    if (ACT == 2) v = fmaxf(v, 0.0f);
    C[(size_t)(m0 + j + half * 8) * N + n0 + l16] = v;
  }
}

// ---------------------------------------------------------------------------
// Persistent LSTM direction: one block of 1024 threads (32 waves).
// Wave n owns gate columns [16n,16n+16). Per step: g = h @ Whh^T via 32
// chained WMMA_F32_16x16x4 (A from LDS h, B from Whh in L2), add precomputed
// input projection xp[b,t,:], then all threads do the elementwise cell update.
// out written into [B, L, 2H] at column offset out_off (0 fwd / HD bwd).
// ---------------------------------------------------------------------------
__global__ __launch_bounds__(1024) void lstm_kernel(
    const float* __restrict__ xp, const float* __restrict__ Whh,
    float* __restrict__ out, int reverse, int out_off) {
  __shared__ float hs[B_][HS_LD];
  __shared__ float cs[B_][HD_];
  __shared__ float gbuf[B_][G_];

  const int tid = threadIdx.x;
  const int lane = tid & 31;
  const int wave = tid >> 5;
  const int l16 = lane & 15;
  const int half = lane >> 4;
  const int n0 = wave * 16;

  const float* brow = Whh + (size_t)(n0 + l16) * HD_ + half * 2;
  const float* hrow = &hs[l16][half * 2];

  for (int i = tid; i < B_ * HS_LD; i += 1024) ((float*)hs)[i] = 0.0f;
  for (int i = tid; i < B_ * HD_; i += 1024) ((float*)cs)[i] = 0.0f;
  __syncthreads();

  for (int step = 0; step < L_; step++) {
    const int t = reverse ? (L_ - 1 - step) : step;

    v8f acc = {0.f, 0.f, 0.f, 0.f, 0.f, 0.f, 0.f, 0.f};
#pragma unroll
    for (int kk = 0; kk < HD_ / 4; kk++) {
      v2f av = *(const v2f*)(hrow + kk * 4);
      v2f bv = *(const v2f*)(brow + kk * 4);
      acc = __builtin_amdgcn_wmma_f32_16x16x4_f32(false, av, false, bv,
                                                  (short)0, acc, false, false);
    }
#pragma unroll
    for (int j = 0; j < 8; j++) {
      const int b = j + half * 8;
      gbuf[b][n0 + l16] = acc[j] + xp[((size_t)b * L_ + t) * G_ + n0 + l16];
    }
    __syncthreads();

    for (int p = tid; p < B_ * HD_; p += 1024) {
      const int b = p >> 7;
      const int hd = p & (HD_ - 1);
      const float ig = gbuf[b][hd];
      const float fg = gbuf[b][HD_ + hd];
      const float gg = gbuf[b][2 * HD_ + hd];
      const float og = gbuf[b][3 * HD_ + hd];
      const float c = sigmf(fg) * cs[b][hd] + sigmf(ig) * tanhf(gg);
      const float h = sigmf(og) * tanhf(c);
      cs[b][hd] = c;
      hs[b][hd] = h;
      out[((size_t)b * L_ + t) * D_ + out_off + hd] = h;
    }
    __syncthreads();
  }
}

// ---------------------------------------------------------------------------
// Self-attention pool: T = tanh(xW^T+b) precomputed [B*L, D].
// s[l] = T[b,l,:].v ; a = softmax_l(s) ; out[b,:] = sum_l a[l] x[b,l,:]
// grid = B, block = L (128 threads).
// ---------------------------------------------------------------------------
__global__ __launch_bounds__(128) void attn_pool(
    const float* __restrict__ T, const float* __restrict__ x,
    const float* __restrict__ v, float* __restrict__ out) {
  __shared__ float sh[L_];
  __shared__ float aw[L_];
  const int b = blockIdx.x;
  const int l = threadIdx.x;

  const float* trow = T + ((size_t)b * L_ + l) * D_;
  float s = 0.0f;
  for (int d = 0; d < D_; d++) s += trow[d] * v[d];
  sh[l] = s;
  __syncthreads();
  for (int off = 64; off > 0; off >>= 1) {
    if (l < off) sh[l] = fmaxf(sh[l], sh[l + off]);
    __syncthreads();
  }
  const float m = sh[0];
  __syncthreads();
  const float e = expf(s - m);
  aw[l] = e;
  sh[l] = e;
  __syncthreads();
  for (int off = 64; off > 0; off >>= 1) {
    if (l < off) sh[l] += sh[l + off];
    __syncthreads();
  }
  const float inv = 1.0f / sh[0];

  float a0 = 0.0f, a1 = 0.0f;
  for (int ll = 0; ll < L_; ll++) {
    const float w = aw[ll] * inv;
    const float* xr = x + ((size_t)b * L_ + ll) * D_;
    a0 += w * xr[l];
    a1 += w * xr[l + L_];
  }
  out[(size_t)b * D_ + l] = a0;
  out[(size_t)b * D_ + l + L_] = a1;
}

// ---------------------------------------------------------------------------
// Co-attention context: e[l] = sum_d tanh(us[b,t,d]+wx[b,l,d]) v[d];
// a = softmax_l(e); ctx[b,t,:] = sum_l a[l] x[b,l,:]
// grid = (L_s, B), block = L_x (128).
// ---------------------------------------------------------------------------
__global__ __launch_bounds__(128) void coattn_ctx(
    const float* __restrict__ wx, const float* __restrict__ us,
    const float* __restrict__ x, const float* __restrict__ v,
    float* __restrict__ ctx) {
  __shared__ float sh[L_];
  __shared__ float aw[L_];
  const int t = blockIdx.x;
  const int b = blockIdx.y;
  const int l = threadIdx.x;

  const float* usr = us + ((size_t)b * L_ + t) * D_;
  const float* wxr = wx + ((size_t)b * L_ + l) * D_;
  float e = 0.0f;
  for (int d = 0; d < D_; d++) e += tanhf(usr[d] + wxr[d]) * v[d];
  sh[l] = e;
  __syncthreads();
  for (int off = 64; off > 0; off >>= 1) {
    if (l < off) sh[l] = fmaxf(sh[l], sh[l + off]);
    __syncthreads();
  }
  const float m = sh[0];
  __syncthreads();
  const float ex = expf(e - m);
  aw[l] = ex;
  sh[l] = ex;
  __syncthreads();
  for (int off = 64; off > 0; off >>= 1) {
    if (l < off) sh[l] += sh[l + off];
    __syncthreads();
  }
  const float inv = 1.0f / sh[0];

  float a0 = 0.0f, a1 = 0.0f;
  for (int ll = 0; ll < L_; ll++) {
    const float w = aw[ll] * inv;
    const float* xr = x + ((size_t)b * L_ + ll) * D_;
    a0 += w * xr[l];
    a1 += w * xr[l + L_];
  }
  ctx[((size_t)b * L_ + t) * D_ + l] = a0;
  ctx[((size_t)b * L_ + t) * D_ + l + L_] = a1;
}

// ---------------------------------------------------------------------------
// Feature concat: [gjj, grr, gjj-grr, gjj*grr, gj, gr, gj-gr, gj*gr] -> [B,8D]
// ---------------------------------------------------------------------------
__global__ __launch_bounds__(256) void concat_kernel(
    const float* __restrict__ gjj, const float* __restrict__ grr,
    const float* __restrict__ gj, const float* __restrict__ gr,
    float* __restrict__ xcat) {
  const int i = blockIdx.x * 256 + threadIdx.x;  // 0 .. B*2048-1
  const int b = i >> 11;
  const int c = i & 2047;
  const int seg = c >> 8;
  const int d = c & (D_ - 1);
  const float a0 = gjj[(size_t)b * D_ + d];
  const float b0 = grr[(size_t)b * D_ + d];
  const float c0 = gj[(size_t)b * D_ + d];
  const float d0 = gr[(size_t)b * D_ + d];
  float val;
  if (seg == 0)      val = a0;
  else if (seg == 1) val = b0;
  else if (seg == 2) val = a0 - b0;
  else if (seg == 3) val = a0 * b0;
  else if (seg == 4) val = c0;
  else if (seg == 5) val = d0;
  else if (seg == 6) val = c0 - d0;
  else               val = c0 * d0;
  xcat[(size_t)b * 2048 + c] = val;
}

// ---------------------------------------------------------------------------
// Final logit: out[b] = sigmoid(h[b,:].W2 + b2). grid = B, block = 256.
// ---------------------------------------------------------------------------
__global__ __launch_bounds__(256) void final_kernel(
    const float* __restrict__ h, const float* __restrict__ W2,
    const float* __restrict__ b2, float* __restrict__ out) {
  __shared__ float sh[256];
  const int b = blockIdx.x;
  const int tid = threadIdx.x;
  float s = 0.0f;
  for (int d = tid; d < 2048; d += 256) s += h[(size_t)b * 2048 + d] * W2[d];
  sh[tid] = s;
  __syncthreads();
  for (int off = 128; off > 0; off >>= 1) {
    if (tid < off) sh[tid] += sh[tid + off];
    __syncthreads();
  }
  if (tid == 0) out[b] = sigmf(sh[0] + b2[0]);
}

// ---------------------------------------------------------------------------

extern "C" void kernel_launch(void* const* d_in, const int* in_sizes, int n_in,
                              void* d_out, int out_size, void* d_ws, size_t ws_size,
                              hipStream_t stream) {
  (void)in_sizes; (void)n_in; (void)out_size; (void)ws_size;
  const float* job  = (const float*)d_in[0];
  const float* user = (const float*)d_in[1];
  // user_lstm
  const float* uWihF = (const float*)d_in[2];
  const float* uWhhF = (const float*)d_in[3];
  const float* ubihF = (const float*)d_in[4];
  const float* ubhhF = (const float*)d_in[5];
  const float* uWihB = (const float*)d_in[6];
  const float* uWhhB = (const float*)d_in[7];
  const float* ubihB = (const float*)d_in[8];
  const float* ubhhB = (const float*)d_in[9];
  // job_lstm
  const float* jWihF = (const float*)d_in[10];
  const float* jWhhF = (const float*)d_in[11];
  const float* jbihF = (const float*)d_in[12];
  const float* jbhhF = (const float*)d_in[13];
  const float* jWihB = (const float*)d_in[14];
  const float* jWhhB = (const float*)d_in[15];
  const float* jbihB = (const float*)d_in[16];
  const float* jbhhB = (const float*)d_in[17];
  // job_sa / user_sa
  const float* jsaW = (const float*)d_in[18];
  const float* jsab = (const float*)d_in[19];
  const float* jsav = (const float*)d_in[20];
  const float* usaW = (const float*)d_in[21];
  const float* usab = (const float*)d_in[22];
  const float* usav = (const float*)d_in[23];
  // job_ca
  const float* jcaW = (const float*)d_in[24];
  const float* jcaU = (const float*)d_in[25];
  const float* jcav = (const float*)d_in[26];
  const float* jcsW = (const float*)d_in[27];
  const float* jcsb = (const float*)d_in[28];
  const float* jcsv = (const float*)d_in[29];
  // user_ca
  const float* ucaW = (const float*)d_in[30];
  const float* ucaU = (const float*)d_in[31];
  const float* ucav = (const float*)d_in[32];
  const float* ucsW = (const float*)d_in[33];
  const float* ucsb = (const float*)d_in[34];
  const float* ucsv = (const float*)d_in[35];
  // mlp
  const float* W1 = (const float*)d_in[36];
  const float* b1 = (const float*)d_in[37];
  const float* W2 = (const float*)d_in[38];
  const float* b2 = (const float*)d_in[39];

  float* ws = (float*)d_ws;
  // workspace layout (floats)
  const size_t XP_UF = 0;                 // [B*L, G]  = 1M each
  const size_t XP_UB = 1048576;
  const size_t XP_JF = 2097152;
  const size_t XP_JB = 3145728;
  const size_t UV    = 4194304;           // [B*L, D] = 512K
  const size_t JV    = 4718592;
  // xp region reused after LSTMs:
  const size_t WX    = XP_UF;             // [B*L, D]
  const size_t US    = XP_UB;             // [B*L, D]
  const size_t CTX   = XP_JF;             // [B*L, D]
  const size_t TBUF  = XP_JB;             // [B*L, D]
  const size_t GJ    = 5242880;           // [B, D]
  const size_t GR    = GJ + 4096;
  const size_t GJJ   = GR + 4096;
  const size_t GRR   = GJJ + 4096;
  const size_t XCAT  = GRR + 4096;        // [B, 8D] = 32768
  const size_t HM    = XCAT + 32768;      // [B, 8D]

  const int ML = B_ * L_;  // 2048 rows
  dim3 blk32(32);

  // ---- Stage A: input projections xp = x @ Wih^T + bih + bhh ----
  dim3 gproj(G_ / 16, ML / 16);
  gemm_wmma<0><<<gproj, blk32, 0, stream>>>(user, uWihF, ubihF, ubhhF, ws + XP_UF, G_, DIN_);
  gemm_wmma<0><<<gproj, blk32, 0, stream>>>(user, uWihB, ubihB, ubhhB, ws + XP_UB, G_, DIN_);
  gemm_wmma<0><<<gproj, blk32, 0, stream>>>(job,  jWihF, jbihF, jbhhF, ws + XP_JF, G_, DIN_);
  gemm_wmma<0><<<gproj, blk32, 0, stream>>>(job,  jWihB, jbihB, jbhhB, ws + XP_JB, G_, DIN_);

  // ---- Stage B: LSTM recurrences (persistent single-WG per direction) ----
  lstm_kernel<<<1, 1024, 0, stream>>>(ws + XP_UF, uWhhF, ws + UV, 0, 0);
  lstm_kernel<<<1, 1024, 0, stream>>>(ws + XP_UB, uWhhB, ws + UV, 1, HD_);
  lstm_kernel<<<1, 1024, 0, stream>>>(ws + XP_JF, jWhhF, ws + JV, 0, 0);
  lstm_kernel<<<1, 1024, 0, stream>>>(ws + XP_JB, jWhhB, ws + JV, 1, HD_);

  // ---- Stage C: self attentions ----
  dim3 gatt(D_ / 16, ML / 16);
  gemm_wmma<1><<<gatt, blk32, 0, stream>>>(ws + JV, jsaW, jsab, nullptr, ws + TBUF, D_, D_);
  attn_pool<<<B_, L_, 0, stream>>>(ws + TBUF, ws + JV, jsav, ws + GJ);
  gemm_wmma<1><<<gatt, blk32, 0, stream>>>(ws + UV, usaW, usab, nullptr, ws + TBUF, D_, D_);
  attn_pool<<<B_, L_, 0, stream>>>(ws + TBUF, ws + UV, usav, ws + GR);

  // ---- Stage D: co-attention (job: x=jv, s=uv) ----
  gemm_wmma<0><<<gatt, blk32, 0, stream>>>(ws + JV, jcaW, nullptr, nullptr, ws + WX, D_, D_);
  gemm_wmma<0><<<gatt, blk32, 0, stream>>>(ws + UV, jcaU, nullptr, nullptr, ws + US, D_, D_);
  coattn_ctx<<<dim3(L_, B_), L_, 0, stream>>>(ws + WX, ws + US, ws + JV, jcav, ws + CTX);
  gemm_wmma<1><<<gatt, blk32, 0, stream>>>(ws + CTX, jcsW, jcsb, nullptr, ws + TBUF, D_, D_);
  attn_pool<<<B_, L_, 0, stream>>>(ws + TBUF, ws + CTX, jcsv, ws + GJJ);

  // ---- co-attention (user: x=uv, s=jv) ----
  gemm_wmma<0><<<gatt, blk32, 0, stream>>>(ws + UV, ucaW, nullptr, nullptr, ws + WX, D_, D_);
  gemm_wmma<0><<<gatt, blk32, 0, stream>>>(ws + JV, ucaU, nullptr, nullptr, ws + US, D_, D_);
  coattn_ctx<<<dim3(L_, B_), L_, 0, stream>>>(ws + WX, ws + US, ws + UV, ucav, ws + CTX);
  gemm_wmma<1><<<gatt, blk32, 0, stream>>>(ws + CTX, ucsW, ucsb, nullptr, ws + TBUF, D_, D_);
  attn_pool<<<B_, L_, 0, stream>>>(ws + TBUF, ws + CTX, ucsv, ws + GRR);

  // ---- Stage E: concat + MLP ----
  concat_kernel<<<(B_ * 2048) / 256, 256, 0, stream>>>(ws + GJJ, ws + GRR, ws + GJ, ws + GR, ws + XCAT);
  gemm_wmma<2><<<dim3(2048 / 16, B_ / 16), blk32, 0, stream>>>(ws + XCAT, W1, b1, nullptr, ws + HM, 2048, 2048);
  final_kernel<<<B_, 256, 0, stream>>>(ws + HM, W2, b2, (float*)d_out);
}